// Model_30193620090924
// MI455X (gfx1250) — compile-verified
//
#include <hip/hip_runtime.h>

typedef __attribute__((ext_vector_type(2))) float v2f;
typedef __attribute__((ext_vector_type(8))) float v8f;

#define C_IN    128
#define H_IN    64
#define W_IN    64
#define C_OUT   256
#define H_OUT   62
#define W_OUT   62
#define BH      8                     // output rows per band
#define ROWS    (BH + 2)              // staged input rows (halo = 2)
#define CSTR    17                    // padded LDS c-stride
#define RSTR    (W_IN * CSTR)         // 1088 floats per staged row
#define LDS_ELEMS (ROWS * RSTR + 64)  // +64 pad: tail half=1 over-read stays in-bounds

// grad_weight[o,c,kh,kw] = sum_{n,i,j} go[n,o,i,j] * x[n,c,i+kh,j+kw]
__global__ __launch_bounds__(128) void conv_bwdw_wmma(
    const float* __restrict__ x, const float* __restrict__ go,
    float* __restrict__ out)
{
    __shared__ float ldsx[LDS_ELEMS];   // [(r*64+col)*17 + c]

    const int tid  = threadIdx.x;
    const int lane = tid & 31;
    const int wave = tid >> 5;       // 0..3
    const int half = lane >> 4;      // 0 or 1
    const int idx  = lane & 15;      // 0..15

    int bid = blockIdx.x;
    const int n  = bid & 31;  bid >>= 5;   // image
    const int ct = bid & 7;   bid >>= 3;   // c-tile
    const int og = bid;                    // 0..3

    const int c0 = ct * 16;
    const int o0 = (og * 4 + wave) * 16;

    const float* xbase = x  + ((size_t)n * C_IN  + c0)       * (H_IN  * W_IN);
    const float* gbase = go + ((size_t)n * C_OUT + o0 + idx) * (H_OUT * W_OUT);

    v8f acc[9];
    #pragma unroll
    for (int t = 0; t < 9; ++t)
        acc[t] = (v8f){0.f, 0.f, 0.f, 0.f, 0.f, 0.f, 0.f, 0.f};

    for (int rbase = 0; rbase < H_OUT; rbase += BH) {
        const int orows = (H_OUT - rbase < BH) ? (H_OUT - rbase) : BH;

        __syncthreads();
        // stage X[c0..c0+15][rbase..rbase+ROWS-1][0..63] -> LDS (fixed trip count;
        // rows past 63 clamp to 63, they are never read)
        #pragma unroll 4
        for (int e = tid; e < 16 * ROWS * W_IN; e += 128) {
            const int col = e & 63;
            const int r   = (e >> 6) % ROWS;          // constant divisor
            const int c   = e / (64 * ROWS);
            int srow = rbase + r;
            srow = (srow > H_IN - 1) ? (H_IN - 1) : srow;
            ldsx[(r * W_IN + col) * CSTR + c] =
                xbase[((size_t)c * H_IN + srow) * W_IN + col];
        }
        __syncthreads();

        for (int ir = 0; ir < orows; ++ir) {
            const float* grow = gbase + (rbase + ir) * W_OUT;
            if (rbase + ir + 1 < H_OUT)
                __builtin_prefetch(grow + W_OUT, 0, 1);   // next row of A

            // ---- main: 15 full K=4 steps, completely branch-free ----
            #pragma unroll 3
            for (int jb = 0; jb < 60; jb += 4) {
                const int j = jb + 2 * half;
                const v2f a = *(const v2f*)(grow + j);        // 8B aligned (j even)
                const int base = (ir * W_IN + j) * CSTR + idx;

                v2f b[9];
                #pragma unroll
                for (int kh = 0; kh < 3; ++kh) {
                    #pragma unroll
                    for (int kw = 0; kw < 3; ++kw) {
                        const int off = base + kh * RSTR + kw * CSTR;
                        b[kh * 3 + kw].x = ldsx[off];
                        b[kh * 3 + kw].y = ldsx[off + CSTR];
                    }
                }
                #pragma unroll
                for (int t = 0; t < 9; ++t)
                    acc[t] = __builtin_amdgcn_wmma_f32_16x16x4_f32(
                        false, a, false, b[t], (short)0, acc[t], false, false);
            }

            // ---- tail: jb=60, only k=60,61 valid (half==0 lanes) ----
            {
                const int  j  = 60 + 2 * half;
                const bool ok = (half == 0);
                const int  ja = ok ? j : 56;                  // clamp: no OOB A load
                const v2f  a  = *(const v2f*)(grow + ja);
                const int base = (ir * W_IN + j) * CSTR + idx; // half=1 lands in pad

                v2f b[9];
                #pragma unroll
                for (int kh = 0; kh < 3; ++kh) {
                    #pragma unroll
                    for (int kw = 0; kw < 3; ++kw) {
                        const int off = base + kh * RSTR + kw * CSTR;
                        const float bx = ldsx[off];
                        const float by = ldsx[off + CSTR];
                        b[kh * 3 + kw].x = ok ? bx : 0.f;     // v_cndmask, no exec ops
                        b[kh * 3 + kw].y = ok ? by : 0.f;
                    }
                }
                #pragma unroll
                for (int t = 0; t < 9; ++t)
                    acc[t] = __builtin_amdgcn_wmma_f32_16x16x4_f32(
                        false, a, false, b[t], (short)0, acc[t], false, false);
            }
        }
    }

    // accumulate split-K (over n) partials into grad_weight
    #pragma unroll
    for (int t = 0; t < 9; ++t) {
        #pragma unroll
        for (int v = 0; v < 8; ++v) {
            const int o = o0 + v + 8 * half;   // D layout: M = v + 8*half
            const int c = c0 + idx;            // D layout: N = lane%16
            atomicAdd(&out[((size_t)o * C_IN + c) * 9 + t], acc[t][v]);
        }
    }
}

__global__ void zero_out_kernel(float* out, int nel) {
    int i = blockIdx.x * blockDim.x + threadIdx.x;
    if (i < nel) out[i] = 0.0f;
}

extern "C" void kernel_launch(void* const* d_in, const int* in_sizes, int n_in,
                              void* d_out, int out_size, void* d_ws, size_t ws_size,
                              hipStream_t stream) {
    const float* x  = (const float*)d_in[0];   // input  [32,128,64,64]
    const float* go = (const float*)d_in[1];   // grad_output [32,256,62,62]
    float* out = (float*)d_out;                // grad_weight [256,128,3,3]

    const int nel = C_OUT * C_IN * 9;          // 294912
    zero_out_kernel<<<(nel + 255) / 256, 256, 0, stream>>>(out, nel);

    // 4 o-groups * 8 c-tiles * 32 images = 1024 blocks of 4 waves
    conv_bwdw_wmma<<<dim3(4 * 8 * 32), 128, 0, stream>>>(x, go, out);
}